// DenSparseMatrix_47210280518072
// MI455X (gfx1250) — compile-verified
//
#include <hip/hip_runtime.h>
#include <stdint.h>

#define IDIM 65536
#define ODIM 65536
#define BDIM 32
#define WDIM 32

// ---------------- Kernel 0: g = forward_weights * forward_mask ----------------
__global__ void k_compute_g(const float4* __restrict__ fw,
                            const float4* __restrict__ fm,
                            float4* __restrict__ g, int n4) {
  int i = blockIdx.x * blockDim.x + threadIdx.x;
  if (i < n4) {
    float4 a = fw[i], b = fm[i];
    g[i] = make_float4(a.x * b.x, a.y * b.y, a.z * b.z, a.w * b.w);
  }
}

// ---------------- Kernel 1: xT[i][b] = x[b][i] (LDS tile transpose) ----------------
__global__ void k_transpose_x(const float* __restrict__ x, float* __restrict__ xT) {
  __shared__ float tile[32][33];
  int i0 = blockIdx.x * 32;
  int tx = threadIdx.x, ty = threadIdx.y;
#pragma unroll
  for (int r = 0; r < 4; ++r) {
    int b = ty + r * 8;
    tile[b][tx] = x[(size_t)b * IDIM + i0 + tx];
  }
  __syncthreads();
#pragma unroll
  for (int r = 0; r < 4; ++r) {
    int il = ty + r * 8;
    xT[(size_t)(i0 + il) * BDIM + tx] = tile[tx][il];
  }
}

// ---------------- Kernel 2: main gather / reduce ----------------
// One wave (32 lanes) per 32-output tile; lane == batch index b.
// map/rm tiles are prefetched to LDS with CDNA5 async-to-LDS DMA, double buffered.

constexpr int WPB = 2;   // waves per block
constexpr int TPW = 2;   // output tiles per wave (double-buffered pipeline)

constexpr unsigned STG_MAP  = 0;      // 2 x 4096 B  (map int32 tile, double buffered)
constexpr unsigned STG_RM   = 8192;   // 2 x 4096 B  (reverse_mask tile, double buffered)
constexpr unsigned MC_OFF   = 16384;  // 1024 x uint2 (m, c) = 8192 B
constexpr unsigned OST_OFF  = 24576;  // 32 x 33 floats = 4224 B (output transpose staging)
constexpr unsigned PER_WAVE = 28800;  // padded per-wave LDS footprint

__device__ __forceinline__ void async_b128(unsigned lds_addr, uint64_t saddr,
                                           unsigned voff) {
  // GVS mode: global addr = SGPR base + 32-bit VGPR offset; VDST VGPR = LDS byte addr.
  asm volatile("global_load_async_to_lds_b128 %0, %1, %2"
               :: "v"(lds_addr), "v"(voff), "s"(saddr)
               : "memory");
}

__global__ __launch_bounds__(WPB * 32) void k_main(
    const int* __restrict__ map, const float* __restrict__ rm,
    const float* __restrict__ g, const float* __restrict__ xT,
    float* __restrict__ out) {
  __shared__ unsigned char lds_raw[WPB * PER_WAVE];

  const unsigned tid  = threadIdx.x;
  const unsigned wave = tid >> 5;
  const unsigned lane = tid & 31;

  unsigned char* wbase = lds_raw + wave * PER_WAVE;
  int*   smap = (int*)(wbase + STG_MAP);
  float* srm  = (float*)(wbase + STG_RM);
  uint2* mc   = (uint2*)(wbase + MC_OFF);
  float* ost  = (float*)(wbase + OST_OFF);

  // Low 32 bits of a generic LDS pointer == LDS byte address (flat->LDS truncation).
  const unsigned lds_wbase = (unsigned)(size_t)(void*)wbase;
  const uint64_t map_sa = (uint64_t)(size_t)map;
  const uint64_t rm_sa  = (uint64_t)(size_t)rm;

  const unsigned gwave  = blockIdx.x * WPB + wave;
  const unsigned o_base = gwave * (32u * TPW);

  // ---- prologue: async prefetch of tile 0 (16 x b128 = 4KB map + 4KB rm) ----
  {
    const unsigned goff = o_base * (WDIM * 4u);  // byte offset (int32 map, f32 rm)
#pragma unroll
    for (int i = 0; i < 8; ++i) {
      unsigned off = (unsigned)(i * 32 + (int)lane) * 16u;
      async_b128(lds_wbase + STG_MAP + off, map_sa, goff + off);
      async_b128(lds_wbase + STG_RM  + off, rm_sa,  goff + off);
    }
  }

  for (int k = 0; k < TPW; ++k) {
    const unsigned o0  = o_base + (unsigned)k * 32u;
    const unsigned buf = (unsigned)k & 1u;

    if (k + 1 < TPW) {
      // issue next tile's async DMA, then drain only the previous 16 ops
      const unsigned goff = (o0 + 32u) * (WDIM * 4u);
      const unsigned lbuf = ((unsigned)(k + 1) & 1u) * 4096u;
#pragma unroll
      for (int i = 0; i < 8; ++i) {
        unsigned off = (unsigned)(i * 32 + (int)lane) * 16u;
        async_b128(lds_wbase + STG_MAP + lbuf + off, map_sa, goff + off);
        async_b128(lds_wbase + STG_RM  + lbuf + off, rm_sa,  goff + off);
      }
      asm volatile("s_wait_asynccnt 16" ::: "memory");
    } else {
      asm volatile("s_wait_asynccnt 0" ::: "memory");
    }

    // ---- Phase B: lane == w; c[o,w] = rm[o,w] * g[map[o,w], w]  ----
#pragma unroll 4
    for (int t = 0; t < 32; ++t) {
      const int idx = (int)buf * 1024 + t * 32 + (int)lane;
      const unsigned m = (unsigned)smap[idx];
      const float cv = srm[idx] * g[(size_t)m * WDIM + lane];
      mc[t * 32 + (int)lane] = make_uint2(m, __float_as_uint(cv));
    }

    // ---- Phase C: lane == b; acc = sum_w c * xT[m, b] (coalesced 128B line/w) ----
    for (int t = 0; t < 32; ++t) {
      float acc = 0.f;
#pragma unroll 8
      for (int w = 0; w < 32; ++w) {
        const uint2 p = mc[t * 32 + w];                    // wave-uniform LDS broadcast
        acc += __uint_as_float(p.y) * xT[(size_t)p.x * BDIM + lane];
      }
      ost[t * 33 + (int)lane] = acc;                       // conflict-free (pad 33)
    }

    // ---- coalesced transposed store: out[b, o0+lane] ----
#pragma unroll 4
    for (int r = 0; r < 32; ++r) {
      out[(size_t)r * ODIM + o0 + lane] = ost[lane * 33u + (unsigned)r];
    }
  }
}

// ---------------- launcher ----------------
extern "C" void kernel_launch(void* const* d_in, const int* in_sizes, int n_in,
                              void* d_out, int out_size, void* d_ws, size_t ws_size,
                              hipStream_t stream) {
  (void)in_sizes; (void)n_in; (void)out_size; (void)ws_size;
  const float* x  = (const float*)d_in[0];
  const float* fw = (const float*)d_in[1];
  const float* fm = (const float*)d_in[2];
  const int*   mp = (const int*)d_in[3];   // output_mapping (int)
  const float* rm = (const float*)d_in[4];
  float* out = (float*)d_out;

  float* xT = (float*)d_ws;                                   // I*32 floats = 8 MB
  float* g  = (float*)((char*)d_ws + (size_t)IDIM * 32 * 4);  // next 8 MB

  const int n4 = IDIM * WDIM / 4;
  k_compute_g<<<(n4 + 255) / 256, 256, 0, stream>>>(
      (const float4*)fw, (const float4*)fm, (float4*)g, n4);

  dim3 tb(32, 8);
  k_transpose_x<<<IDIM / 32, tb, 0, stream>>>(x, xT);

  const int blocks = ODIM / (32 * TPW * WPB);   // 512
  k_main<<<blocks, WPB * 32, 0, stream>>>(mp, rm, g, xT, out);
}